// WrappedOriginalGCN_38560216383500
// MI455X (gfx1250) — compile-verified
//
#include <hip/hip_runtime.h>

// ---------------------------------------------------------------------------
// GCN on MI455X (gfx1250, wave32). Sparse reformulation of the dense reference:
//   A = D^-1/2 (scatter_sym(w/2) + I) D^-1/2
//   A @ y  ==  dis * ( u + edge_scatter(u) ),  u = dis * y
// Dense GEMMs (x@W1, h@W2) use V_WMMA_F32_16X16X4_F32 (full fp32 precision).
// Edge scatters are f32 atomics into L2-resident (<=4MB) accumulators.
// ---------------------------------------------------------------------------

typedef __attribute__((ext_vector_type(2))) float v2f;
typedef __attribute__((ext_vector_type(8))) float v8f;

// ---------------- degree / normalization ----------------

__global__ void init_deg_kernel(float* __restrict__ deg, int N) {
    int i = blockIdx.x * blockDim.x + threadIdx.x;
    if (i < N) deg[i] = 1.0f;  // self-loop from +I
}

__global__ void deg_scatter_kernel(const int* __restrict__ ei,
                                   const float* __restrict__ w,
                                   float* __restrict__ deg, int E) {
    int e = blockIdx.x * blockDim.x + threadIdx.x;
    if (e >= E) return;
    int r = ei[e];
    int c = ei[E + e];
    float hw = 0.5f * w[e];
    atomicAdd(&deg[r], hw);
    atomicAdd(&deg[c], hw);
}

__global__ void make_dis_kernel(const float* __restrict__ deg,
                                float* __restrict__ dis, int N) {
    int i = blockIdx.x * blockDim.x + threadIdx.x;
    if (i < N) {
        float d = deg[i];
        dis[i] = (d > 0.0f) ? rsqrtf(d) : 0.0f;
    }
}

// ---------------- fp32 WMMA GEMM with fused dis-scale epilogue ----------------
// Y = X[M x K] @ B[K x NCOL];  U = dis*Y row-scaled;  Z initialized = U (self loop).
// One wave per 16x16 output tile, K-loop in steps of 4 via v_wmma_f32_16x16x4_f32.

template <int K, int NCOL>
__global__ __launch_bounds__(256)
void gemm_wmma_scale_kernel(const float* __restrict__ X,
                            const float* __restrict__ B,
                            const float* __restrict__ dis,
                            float* __restrict__ U,
                            float* __restrict__ Z,
                            int M) {
    const int ntiles = NCOL / 16;
    const int wave   = (int)((blockIdx.x * blockDim.x + threadIdx.x) >> 5);
    const int lane   = threadIdx.x & 31;
    const int mtiles = M / 16;
    if (wave >= mtiles * ntiles) return;  // wave-uniform: EXEC all-1 inside

    const int m0   = (wave / ntiles) * 16;
    const int n0   = (wave % ntiles) * 16;
    const int half = lane >> 4;   // 0: K pair {k0,k0+1}; 1: K pair {k0+2,k0+3}
    const int l15  = lane & 15;

    const int arow = m0 + l15;    // A-frag row (M striped on lanes 0..15 / 16..31)
    const int n    = n0 + l15;    // B/C/D column striped across lanes

    v8f c = {};
    #pragma unroll
    for (int k0 = 0; k0 < K; k0 += 4) {
        const int kb = k0 + half * 2;
        v2f a;
        a.x = X[arow * K + kb];
        a.y = X[arow * K + kb + 1];
        v2f b;
        b.x = B[kb * NCOL + n];
        b.y = B[(kb + 1) * NCOL + n];
        c = __builtin_amdgcn_wmma_f32_16x16x4_f32(
                /*neg_a=*/false, a, /*neg_b=*/false, b,
                /*c_mod=*/(short)0, c, /*reuse_a=*/false, /*reuse_b=*/false);
    }

    // Epilogue: u = dis[row]*y ; z = u  (self-loop init for the scatter pass)
    #pragma unroll
    for (int r = 0; r < 8; ++r) {
        const int row = m0 + r + half * 8;  // C/D layout: VGPR r -> M=r (+8 for hi lanes)
        const float v = c[r] * dis[row];
        U[row * NCOL + n] = v;
        Z[row * NCOL + n] = v;
    }
}

// ---------------- edge scatter (SpMM), D = 64 : one wave per edge ----------------

__global__ __launch_bounds__(256)
void spmm_scatter64_kernel(const int* __restrict__ ei,
                           const float* __restrict__ w,
                           const float* __restrict__ U,
                           float* __restrict__ Z, int E) {
    const int gid  = blockIdx.x * blockDim.x + threadIdx.x;
    const int e    = gid >> 5;
    const int lane = threadIdx.x & 31;
    if (e >= E) return;
    const int r  = ei[e];
    const int c  = ei[E + e];
    const float hw = 0.5f * w[e];
    const int col = lane * 2;           // 2 contiguous cols per lane (64 total)
    const float2 uc = *(const float2*)&U[c * 64 + col];
    const float2 ur = *(const float2*)&U[r * 64 + col];
    atomicAdd(&Z[r * 64 + col],     hw * uc.x);
    atomicAdd(&Z[r * 64 + col + 1], hw * uc.y);
    atomicAdd(&Z[c * 64 + col],     hw * ur.x);
    atomicAdd(&Z[c * 64 + col + 1], hw * ur.y);
}

// ---------------- edge scatter (SpMM), D = 16 : 16 threads per edge ----------------

__global__ __launch_bounds__(256)
void spmm_scatter16_kernel(const int* __restrict__ ei,
                           const float* __restrict__ w,
                           const float* __restrict__ U,
                           float* __restrict__ Z, int E) {
    const long long gid = (long long)blockIdx.x * blockDim.x + threadIdx.x;
    const int e   = (int)(gid >> 4);
    const int col = (int)(gid & 15);
    if (e >= E) return;
    const int r = ei[e];
    const int c = ei[E + e];
    const float hw = 0.5f * w[e];
    atomicAdd(&Z[r * 16 + col], hw * U[c * 16 + col]);
    atomicAdd(&Z[c * 16 + col], hw * U[r * 16 + col]);
}

// ---------------- finalize layer 1: h = relu(dis*z + b1) ----------------

__global__ void relu_bias_kernel(const float* __restrict__ Z,
                                 const float* __restrict__ dis,
                                 const float* __restrict__ b,
                                 float* __restrict__ H, int total) {
    int i = blockIdx.x * blockDim.x + threadIdx.x;
    if (i >= total) return;
    const int row = i >> 6;
    const int col = i & 63;
    const float v = dis[row] * Z[i] + b[col];
    H[i] = v > 0.0f ? v : 0.0f;
}

// ---------------- finalize layer 2: out = dis*z + b2 ----------------

__global__ void out_bias_kernel(const float* __restrict__ Z,
                                const float* __restrict__ dis,
                                const float* __restrict__ b,
                                float* __restrict__ O, int total) {
    int i = blockIdx.x * blockDim.x + threadIdx.x;
    if (i >= total) return;
    const int row = i >> 4;
    const int col = i & 15;
    O[i] = dis[row] * Z[i] + b[col];
}

// ---------------------------------------------------------------------------

static inline size_t align256(size_t x) { return (x + 255) & ~(size_t)255; }

extern "C" void kernel_launch(void* const* d_in, const int* in_sizes, int n_in,
                              void* d_out, int out_size, void* d_ws, size_t ws_size,
                              hipStream_t stream) {
    const float* x  = (const float*)d_in[0];  // [N, 64]
    const float* W1 = (const float*)d_in[1];  // [64, 64]
    const float* b1 = (const float*)d_in[2];  // [64]
    const float* W2 = (const float*)d_in[3];  // [64, 16]
    const float* b2 = (const float*)d_in[4];  // [16]
    const float* ew = (const float*)d_in[5];  // [E]
    const int*   ei = (const int*)d_in[6];    // [2, E]

    const int N = in_sizes[0] / 64;
    const int E = in_sizes[5];

    // Workspace layout (all fully rewritten each call -> replay-safe)
    char* ws = (char*)d_ws;
    size_t off = 0;
    float* deg = (float*)(ws + off); off += align256((size_t)N * 4);
    float* dis = (float*)(ws + off); off += align256((size_t)N * 4);
    float* u1  = (float*)(ws + off); off += align256((size_t)N * 64 * 4);
    float* z1  = (float*)(ws + off); off += align256((size_t)N * 64 * 4);
    float* h   = (float*)(ws + off); off += align256((size_t)N * 64 * 4);
    float* u2  = (float*)(ws + off); off += align256((size_t)N * 16 * 4);
    float* z2  = (float*)(ws + off); off += align256((size_t)N * 16 * 4);
    (void)ws_size; (void)n_in; (void)out_size;

    float* out = (float*)d_out;  // [N, 16] fp32

    const int TB = 256;

    // 1) degree + symmetric normalization vector
    init_deg_kernel<<<(N + TB - 1) / TB, TB, 0, stream>>>(deg, N);
    deg_scatter_kernel<<<(E + TB - 1) / TB, TB, 0, stream>>>(ei, ew, deg, E);
    make_dis_kernel<<<(N + TB - 1) / TB, TB, 0, stream>>>(deg, dis, N);

    // 2) layer 1: y1 = x@W1 ; u1 = dis*y1 ; z1 = u1 (self loop)
    {
        const int tiles  = (N / 16) * (64 / 16);
        const int blocks = (tiles * 32 + TB - 1) / TB;
        gemm_wmma_scale_kernel<64, 64><<<blocks, TB, 0, stream>>>(x, W1, dis, u1, z1, N);
    }
    // 3) z1 += symmetric edge scatter of u1 (L2-resident atomics)
    {
        const long long threads = (long long)E * 32;
        const int blocks = (int)((threads + TB - 1) / TB);
        spmm_scatter64_kernel<<<blocks, TB, 0, stream>>>(ei, ew, u1, z1, E);
    }
    // 4) h = relu(dis*z1 + b1)
    relu_bias_kernel<<<(N * 64 + TB - 1) / TB, TB, 0, stream>>>(z1, dis, b1, h, N * 64);

    // 5) layer 2: y2 = h@W2 ; u2 = dis*y2 ; z2 = u2
    {
        const int tiles  = (N / 16) * (16 / 16);
        const int blocks = (tiles * 32 + TB - 1) / TB;
        gemm_wmma_scale_kernel<64, 16><<<blocks, TB, 0, stream>>>(h, W2, dis, u2, z2, N);
    }
    // 6) z2 += symmetric edge scatter of u2
    {
        const long long threads = (long long)E * 16;
        const int blocks = (int)((threads + TB - 1) / TB);
        spmm_scatter16_kernel<<<blocks, TB, 0, stream>>>(ei, ew, u2, z2, E);
    }
    // 7) out = dis*z2 + b2
    out_bias_kernel<<<(N * 16 + TB - 1) / TB, TB, 0, stream>>>(z2, dis, b2, out, N * 16);
}